// DisplaceChannel_88656714924642
// MI455X (gfx1250) — compile-verified
//
#include <hip/hip_runtime.h>

// ---------------------------------------------------------------------------
// DisplaceChannel + Gaussian-interp depthwise 5x5 conv for MI455X (gfx1250).
//
// out[b,c,y,x] = sum_{i,j} ky[i]*kx[j] * in[b,c, y+i-2-dy, x+j-2-dx]
// (zero padded), (dx,dy)=round(offset[p]); kx/ky separably-normalized 1-D
// Gaussians of the fractional offset.  Memory-bound (~402 MB @ 23.3 TB/s
// ~= 17us floor): one HBM read per input element, integer shift folded into
// staging addresses (fully shifted-out regions never read), separable conv
// on the matrix unit via V_WMMA_F32_16X16X4_F32 (banded 20x16 tap matrix in
// five K=4 chunks -> 5 wmma per 16x16 tile per direction).  Staging uses the
// gfx1250 async global->LDS path (ASYNCcnt) when available.
// ---------------------------------------------------------------------------

#if defined(__HIP_DEVICE_COMPILE__) && \
    !__has_builtin(__builtin_amdgcn_wmma_f32_16x16x4_f32)
#error "missing __builtin_amdgcn_wmma_f32_16x16x4_f32 for gfx1250"
#endif

#if defined(__HIP_DEVICE_COMPILE__) && \
    __has_builtin(__builtin_amdgcn_global_load_async_to_lds_b32)
#define USE_ASYNC_LDS 1
#else
#define USE_ASYNC_LDS 0
#endif

typedef __attribute__((ext_vector_type(2))) float v2f;
typedef __attribute__((ext_vector_type(8))) float v8f;

#if USE_ASYNC_LDS
typedef __attribute__((address_space(1))) int gint_t;
typedef __attribute__((address_space(3))) int lint_t;
#endif

#define IMG_H 64
#define IMG_W 64
#define NCH   384          // 48 positions * 8 chan/pos
#define SROWS 80           // 5 row-tiles of 16 (covers 68 live rows + zeros)
#define SLD   73           // odd LDS row stride (dwords) -> conflict-free A loads

__device__ __forceinline__ float pick5(float k0, float k1, float k2,
                                       float k3, float k4, int i) {
  float r = 0.0f;
  r = (i == 0) ? k0 : r;
  r = (i == 1) ? k1 : r;
  r = (i == 2) ? k2 : r;
  r = (i == 3) ? k3 : r;
  r = (i == 4) ? k4 : r;
  return r;
}

__global__ __launch_bounds__(256)
void displace_gauss_wmma(const float* __restrict__ x,
                         const float* __restrict__ offset,
                         float* __restrict__ out) {
  __shared__ float s_in [SROWS][SLD];   // zero-haloed, integer-shifted image
  __shared__ float s_tmp[SROWS][SLD];   // horizontal-pass result rows [-2,78)

  const int tid = threadIdx.x;
  const int img = blockIdx.x;           // flat (b*384 + c)
  const int c   = img % NCH;
  const int p   = c >> 3;               // position group (8 channels each)

  // --- per-position separable Gaussian taps -------------------------------
  const float ox = offset[2 * p + 0];
  const float oy = offset[2 * p + 1];
  const float rx = rintf(ox), ry = rintf(oy);   // matches jnp.round
  const int   dx = (int)rx,   dy = (int)ry;
  const float subx = ox - rx, suby = oy - ry;

  const float inv2s2 = 2.0f;            // 1/(2*0.5^2)
  float kx_[5], ky_[5];
  float sx = 0.0f, sy = 0.0f;
#pragma unroll
  for (int t = 0; t < 5; ++t) {
    float fx = (float)(t - 2) + subx;
    float fy = (float)(t - 2) + suby;
    kx_[t] = expf(-fx * fx * inv2s2);
    ky_[t] = expf(-fy * fy * inv2s2);
    sx += kx_[t];
    sy += ky_[t];
  }
  const float rsx = 1.0f / sx, rsy = 1.0f / sy;
#pragma unroll
  for (int t = 0; t < 5; ++t) { kx_[t] *= rsx; ky_[t] *= rsy; }

  // --- stage shifted image into LDS (zero halo, one HBM read per element) --
  {
    float* sflat = &s_in[0][0];
    for (int i = tid; i < SROWS * SLD; i += 256) sflat[i] = 0.0f;
  }
  __syncthreads();

  const float* __restrict__ src = x + (size_t)img * (IMG_H * IMG_W);
  // s_in[yy][xx] = in[yy-2-dy][xx-2-dx]; only the overlap rectangle is read.
  {
    const int ylo = max(2 + dy, 0), yhi = min(66 + dy, 68);
    const int xlo = max(2 + dx, 0), xhi = min(66 + dx, 68);
    const int rseg = tid >> 6;          // 0..3
    const int cseg = tid & 63;          // 0..63
    const int xx = xlo + cseg;
    if (xx < xhi) {
      const int xg = xx - 2 - dx;       // in [0,64)
      for (int yy = ylo + rseg; yy < yhi; yy += 4) {
        const int yg = yy - 2 - dy;     // in [0,64)
#if USE_ASYNC_LDS
        // global_load_async_to_lds_b32: LDS written straight from the memory
        // pipe, tracked by ASYNCcnt (no VGPR round trip).
        gint_t* gp = (gint_t*)(unsigned long long)(uintptr_t)
                         (src + yg * IMG_W + xg);
        lint_t* lp = (lint_t*)(unsigned int)(uintptr_t)(&s_in[yy][xx]);
        __builtin_amdgcn_global_load_async_to_lds_b32(gp, lp, 0, 0);
#else
        s_in[yy][xx] = src[yg * IMG_W + xg];
#endif
      }
    }
  }
#if USE_ASYNC_LDS
  asm volatile("s_wait_asynccnt 0" ::: "memory");
#endif
  __syncthreads();

  // --- wave32 WMMA operand lane mapping (ISA 7.12.2) ----------------------
  const int lane = tid & 31;
  const int wid  = tid >> 5;            // 8 waves
  const int half = lane >> 4;           // 0 | 1
  const int lm   = lane & 15;           // M (A/C) or N (B/C) index

  // Banded tap matrices: value = tap[K - idx], K = 4k + 2*half + {0,1}.
  v2f bx[5], ay[5];
#pragma unroll
  for (int k = 0; k < 5; ++k) {
    const int i0 = 4 * k + 2 * half - lm;
    bx[k].x = pick5(kx_[0], kx_[1], kx_[2], kx_[3], kx_[4], i0);
    bx[k].y = pick5(kx_[0], kx_[1], kx_[2], kx_[3], kx_[4], i0 + 1);
    ay[k].x = pick5(ky_[0], ky_[1], ky_[2], ky_[3], ky_[4], i0);
    ay[k].y = pick5(ky_[0], ky_[1], ky_[2], ky_[3], ky_[4], i0 + 1);
  }

  // --- pass 1: horizontal 5-tap conv, 20 tiles of 16x16 -------------------
  // out_h rows span [-2,78) stored at s_tmp[row+2]; tile row base rb=16*rt.
  for (int tt = wid; tt < 20; tt += 8) {
    const int rb = (tt >> 2) * 16;      // s_in / s_tmp row base
    const int x0 = (tt & 3) * 16;
    v8f acc = {};
#pragma unroll
    for (int k = 0; k < 5; ++k) {
      const float* ap = &s_in[rb + lm][x0 + 4 * k + 2 * half];
      v2f a;
      a.x = ap[0];
      a.y = ap[1];
      acc = __builtin_amdgcn_wmma_f32_16x16x4_f32(
          false, a, false, bx[k], (short)0, acc, false, false);
    }
#pragma unroll
    for (int v = 0; v < 8; ++v)
      s_tmp[rb + 8 * half + v][x0 + lm] = acc[v];
  }
  __syncthreads();

  // --- pass 2: vertical 5-tap conv, 16 output tiles, stream to HBM --------
  float* __restrict__ dst = out + (size_t)img * (IMG_H * IMG_W);
  for (int tt = wid; tt < 16; tt += 8) {
    const int y0 = (tt >> 2) * 16;
    const int x0 = (tt & 3) * 16;
    v8f acc = {};
#pragma unroll
    for (int k = 0; k < 5; ++k) {
      const int rr = y0 + 4 * k + 2 * half;
      v2f b;
      b.x = s_tmp[rr][x0 + lm];
      b.y = s_tmp[rr + 1][x0 + lm];
      acc = __builtin_amdgcn_wmma_f32_16x16x4_f32(
          false, ay[k], false, b, (short)0, acc, false, false);
    }
#pragma unroll
    for (int v = 0; v < 8; ++v)
      dst[(y0 + 8 * half + v) * IMG_W + (x0 + lm)] = acc[v];
  }
}

extern "C" void kernel_launch(void* const* d_in, const int* in_sizes, int n_in,
                              void* d_out, int out_size, void* d_ws,
                              size_t ws_size, hipStream_t stream) {
  const float* x      = (const float*)d_in[0];   // (32,384,64,64) f32
  const float* offset = (const float*)d_in[1];   // (48,2) f32
  float* out = (float*)d_out;                    // (32,384,64,64) f32

  const int nimg = out_size / (IMG_H * IMG_W);   // 32*384 = 12288
  displace_gauss_wmma<<<nimg, 256, 0, stream>>>(x, offset, out);
}